// GraphAttentionLayer_31782757990557
// MI455X (gfx1250) — compile-verified
//
#include <hip/hip_runtime.h>
#include <math.h>

typedef __attribute__((ext_vector_type(2))) float v2f;
typedef __attribute__((ext_vector_type(8))) float v8f;

#define ALPHA_LRELU 0.2f
#define NEG_INF_GAT -9000000000000000.0f
#define N_NODES 2048
#define FIN 128
#define FOUT 64
#define BATCH 8

// -----------------------------------------------------------------------------
// Kernel A: Wh = h @ W  (B*N=16384 rows, K=128, Fout=64)  +  si = Wh@a1, sj = Wh@a2
// One wave owns a 16-row x 64-col tile: 4 f32 WMMA accumulators, K-loop step 4.
// Epilogue: per-row dot with a1/a2 reduced across 16 lanes via shfl_xor (wave32).
// -----------------------------------------------------------------------------
__global__ __launch_bounds__(256) void gat_wh_kernel(
    const float* __restrict__ h, const float* __restrict__ W,
    const float* __restrict__ a, float* __restrict__ Wh,
    float* __restrict__ si, float* __restrict__ sj)
{
  const int lane = threadIdx.x & 31;
  const int wave = threadIdx.x >> 5;
  const int tile = blockIdx.x * 8 + wave;   // 0..1023 (16384/16 row tiles)
  const int row0 = tile * 16;
  const int half = lane >> 4;               // K sub-select for A/B fragments
  const int l16  = lane & 15;               // M for A, N for B/C/D

  v8f acc0 = {}, acc1 = {}, acc2 = {}, acc3 = {};
  const float* hrow = h + (size_t)(row0 + l16) * FIN;

  for (int k = 0; k < FIN; k += 4) {
    const int kb = k + 2 * half;
    v2f av = *(const v2f*)(hrow + kb);              // A: 16x4 tile, lane M=l16, K=kb,kb+1
    const float* Wr0 = W + (size_t)kb * FOUT;       // B: 4x16 tiles, lane N=l16
    const float* Wr1 = Wr0 + FOUT;
    v2f b0, b1, b2, b3;
    b0.x = Wr0[ 0 + l16]; b0.y = Wr1[ 0 + l16];
    b1.x = Wr0[16 + l16]; b1.y = Wr1[16 + l16];
    b2.x = Wr0[32 + l16]; b2.y = Wr1[32 + l16];
    b3.x = Wr0[48 + l16]; b3.y = Wr1[48 + l16];
    acc0 = __builtin_amdgcn_wmma_f32_16x16x4_f32(false, av, false, b0, (short)0, acc0, false, false);
    acc1 = __builtin_amdgcn_wmma_f32_16x16x4_f32(false, av, false, b1, (short)0, acc1, false, false);
    acc2 = __builtin_amdgcn_wmma_f32_16x16x4_f32(false, av, false, b2, (short)0, acc2, false, false);
    acc3 = __builtin_amdgcn_wmma_f32_16x16x4_f32(false, av, false, b3, (short)0, acc3, false, false);
  }

  float a1f[4], a2f[4];
#pragma unroll
  for (int ft = 0; ft < 4; ++ft) {
    a1f[ft] = a[ft * 16 + l16];
    a2f[ft] = a[FOUT + ft * 16 + l16];
  }

#pragma unroll
  for (int v = 0; v < 8; ++v) {
    const int row = row0 + v + 8 * half;            // C/D layout: VGPR v -> M = v + 8*(lane/16)
    float* wr = Wh + (size_t)row * FOUT;
    wr[ 0 + l16] = acc0[v];
    wr[16 + l16] = acc1[v];
    wr[32 + l16] = acc2[v];
    wr[48 + l16] = acc3[v];
    float p1 = acc0[v]*a1f[0] + acc1[v]*a1f[1] + acc2[v]*a1f[2] + acc3[v]*a1f[3];
    float p2 = acc0[v]*a2f[0] + acc1[v]*a2f[1] + acc2[v]*a2f[2] + acc3[v]*a2f[3];
#pragma unroll
    for (int m = 1; m < 16; m <<= 1) {              // reduce across the 16 lanes sharing row M
      p1 += __shfl_xor(p1, m, 32);
      p2 += __shfl_xor(p2, m, 32);
    }
    if (l16 == 0) { si[row] = p1; sj[row] = p2; }
  }
}

// -----------------------------------------------------------------------------
// Kernel B: fused masked-leakyReLU softmax + attention output + h_prime = att@Wh
// One 128-thread (4-wave) block per (batch, 16-row tile).
//   pass 1: online (max,sum) softmax statistics per row
//   chunk loop (64 j at a time): compute att, write to HBM, stage in LDS,
//     stage Wh 64x64 chunk in LDS, 4 waves each own a 16x16 f-tile of h_prime
//     and accumulate with v_wmma_f32_16x16x4_f32 (16 WMMA per chunk per wave).
//   epilogue: elu and store. Attention is never re-read from HBM.
// -----------------------------------------------------------------------------
__global__ __launch_bounds__(128) void gat_attn_kernel(
    const int* __restrict__ adj, const float* __restrict__ Wh,
    const float* __restrict__ si, const float* __restrict__ sj,
    float* __restrict__ out_h, float* __restrict__ att_out)
{
  __shared__ float sjS[N_NODES];       // 8 KB
  __shared__ float attT[16][64];       // 4 KB  : attention chunk (A matrix)
  __shared__ float whT[64][64];        // 16 KB : Wh chunk (B matrix)
  __shared__ float pm[128], ps[128];   // online-softmax partials
  __shared__ float rowm[16], rinv[16], siS[16];

  const int b  = blockIdx.x / (N_NODES / 16);
  const int i0 = (blockIdx.x % (N_NODES / 16)) * 16;
  const int t  = threadIdx.x;
  const int lane = t & 31, wave = t >> 5;
  const int halfw = lane >> 4, l16 = lane & 15;

  for (int j = t; j < N_NODES; j += 128) sjS[j] = sj[(size_t)b * N_NODES + j];
  if (t < 16) siS[t] = si[(size_t)b * N_NODES + i0 + t];
  __syncthreads();

  // ---- pass 1: online max/sum, 8 threads per row ----
  {
    const int r = t >> 3, c = t & 7;
    const int* arow = adj + (size_t)(i0 + r) * N_NODES;
    const float sir = siS[r];
    float m = -__builtin_inff(), s = 0.f;
    for (int j = c; j < N_NODES; j += 8) {
      float e = sir + sjS[j];
      e = (e > 0.f) ? e : ALPHA_LRELU * e;
      float v = (arow[j] > 0) ? e : NEG_INF_GAT;
      float nm = fmaxf(m, v);
      s = s * __expf(m - nm) + __expf(v - nm);
      m = nm;
    }
    pm[t] = m; ps[t] = s;
  }
  __syncthreads();
  if (t < 16) {
    float m = -__builtin_inff(), s = 0.f;
#pragma unroll
    for (int c = 0; c < 8; ++c) {
      float m2 = pm[t * 8 + c], s2 = ps[t * 8 + c];
      float nm = fmaxf(m, m2);
      s = s * __expf(m - nm) + s2 * __expf(m2 - nm);
      m = nm;
    }
    rowm[t] = m; rinv[t] = 1.f / s;   // s >= 1 always (max element contributes 1)
  }
  __syncthreads();

  // ---- chunk loop: attention values + WMMA accumulate ----
  v8f acc = {};
  const int jl = t & 63;
  const int rg = t >> 6;                                  // 0..1 -> rows rg*8..rg*8+7
  const float* WhB = Wh + (size_t)b * N_NODES * FOUT;
  const int fcol = wave * 16 + l16;                       // this wave's f-tile column

  for (int j0 = 0; j0 < N_NODES; j0 += 64) {
    // stage Wh[j0..j0+63][0..63] via float4 (1024 float4 / 128 threads)
#pragma unroll
    for (int i = 0; i < 8; ++i) {
      int lin = t + 128 * i;
      int jj = lin >> 4, f4 = (lin & 15) * 4;
      *(float4*)&whT[jj][f4] = *(const float4*)(WhB + (size_t)(j0 + jj) * FOUT + f4);
    }
    // attention values for this chunk: 16 rows x 64 cols
#pragma unroll
    for (int q = 0; q < 8; ++q) {
      int r = rg * 8 + q;
      int j = j0 + jl;
      float e = siS[r] + sjS[j];
      e = (e > 0.f) ? e : ALPHA_LRELU * e;
      float v = (adj[(size_t)(i0 + r) * N_NODES + j] > 0) ? e : NEG_INF_GAT;
      float atv = __expf(v - rowm[r]) * rinv[r];
      attT[r][jl] = atv;
      att_out[(size_t)(b * N_NODES + i0 + r) * N_NODES + j] = atv;
    }
    __syncthreads();
    // 4 waves x 16 WMMA: h_prime(16x16 per wave) += att(16x4) @ Wh(4x16)
#pragma unroll
    for (int k4 = 0; k4 < 64; k4 += 4) {
      v2f av = *(const v2f*)&attT[l16][k4 + 2 * halfw];   // A: M=l16, K=kb,kb+1
      v2f bv;
      bv.x = whT[k4 + 2 * halfw][fcol];                   // B: K rows, N=fcol
      bv.y = whT[k4 + 2 * halfw + 1][fcol];
      acc = __builtin_amdgcn_wmma_f32_16x16x4_f32(false, av, false, bv, (short)0, acc, false, false);
    }
    __syncthreads();
  }

  // ---- epilogue: elu + store h_prime tile ----
#pragma unroll
  for (int v = 0; v < 8; ++v) {
    int row = i0 + v + 8 * halfw;
    float x = acc[v];
    float y = (x > 0.f) ? x : expm1f(x);
    out_h[(size_t)(b * N_NODES + row) * FOUT + fcol] = y;
  }
}

// -----------------------------------------------------------------------------
// Launch. Inputs: h(8,2048,128) f32 | adj(2048,2048) i32 | W(128,64) f32 | a(128,1) f32
// Output: elu(h_prime) (8,2048,64) then attention (8,2048,2048), concatenated.
// Workspace: Wh 4MB + si/sj 128KB.
// -----------------------------------------------------------------------------
extern "C" void kernel_launch(void* const* d_in, const int* in_sizes, int n_in,
                              void* d_out, int out_size, void* d_ws, size_t ws_size,
                              hipStream_t stream) {
  const float* h   = (const float*)d_in[0];
  const int*   adj = (const int*)d_in[1];
  const float* W   = (const float*)d_in[2];
  const float* a   = (const float*)d_in[3];

  float* Wh = (float*)d_ws;                          // 16384*64 floats
  float* si = Wh + (size_t)BATCH * N_NODES * FOUT;   // 16384
  float* sj = si + (size_t)BATCH * N_NODES;          // 16384

  float* out_h = (float*)d_out;                                  // 8*2048*64
  float* att   = out_h + (size_t)BATCH * N_NODES * FOUT;         // 8*2048*2048

  gat_wh_kernel<<<128, 256, 0, stream>>>(h, W, a, Wh, si, sj);
  gat_attn_kernel<<<BATCH * (N_NODES / 16), 128, 0, stream>>>(adj, Wh, si, sj, out_h, att);
}